// DTWLoss2_32701880992226
// MI455X (gfx1250) — compile-verified
//
#include <hip/hip_runtime.h>
#include <math.h>

// DTW loss: source (M), target (N), both f32; output = sqrt(DTW[M][N]) (1 f32).
// Anti-diagonal wavefront DP in a single 1024-thread workgroup on one WGP.
// Per diagonal k, cell (i, j=k-i):
//   up      = D[i-1, j]   = neighbor's value from diag k-1  -> LDS read
//   left    = D[i, j-1]   = my value from diag k-1          -> register
//   upleft  = D[i-1, j-1] = neighbor's value from diag k-2  -> register (cached
//                           from last iteration's LDS read)

#define DTW_M 2048
#define DTW_N 2048
#define NTHREADS 1024
#define NROWS (DTW_M + 1)   // rows i = 0..M on a diagonal
#define INFV 1e30f

#if defined(__HIP_DEVICE_COMPILE__) && __has_builtin(__builtin_amdgcn_global_load_async_to_lds_b32)
#define HAVE_ASYNC_LDS 1
#endif

__global__ __launch_bounds__(NTHREADS) void
dtw_wavefront_kernel(const float* __restrict__ src,
                     const float* __restrict__ tgt,
                     float* __restrict__ out)
{
    __shared__ float dbuf[2][NROWS];  // double-buffered anti-diagonals, indexed by row i
    __shared__ float lt[DTW_N];       // target vector in LDS

    const int tid = threadIdx.x;

    // ---- preload target into LDS (CDNA5 async global->LDS path if available) ----
#if defined(HAVE_ASYNC_LDS)
    {
        typedef __attribute__((address_space(1))) int gi_t;  // global int
        typedef __attribute__((address_space(3))) int li_t;  // LDS int
        __builtin_amdgcn_global_load_async_to_lds_b32(
            (gi_t*)(tgt + tid), (li_t*)&lt[tid], 0, 0);
        __builtin_amdgcn_global_load_async_to_lds_b32(
            (gi_t*)(tgt + tid + NTHREADS), (li_t*)&lt[tid + NTHREADS], 0, 0);
#if __has_builtin(__builtin_amdgcn_s_wait_asynccnt)
        __builtin_amdgcn_s_wait_asynccnt(0);
#else
        asm volatile("s_wait_asynccnt 0" ::: "memory");
#endif
    }
#else
    lt[tid]            = tgt[tid];
    lt[tid + NTHREADS] = tgt[tid + NTHREADS];
#endif

    // ---- per-cell register state (thread handles rows tid, tid+1024, tid+2048) ----
    float my_prev[3];   // V(k-1, i)      : my value on previous diagonal
    float nbr_prev[3];  // V(k-2, i-1)    : neighbor's value two diagonals back
    float ss[3];        // source[i-1]
    int   rows[3];

#pragma unroll
    for (int c = 0; c < 3; ++c) {
        const int r = tid + c * NTHREADS;
        rows[c]     = r;
        ss[c]       = (r >= 1 && r <= DTW_M) ? src[r - 1] : 0.0f;
        my_prev[c]  = INFV;
        nbr_prev[c] = INFV;
    }

    float result = INFV;

    // ---- wavefront over diagonals k = i + j, k = 0 .. M+N ----
    for (int k = 0; k <= DTW_M + DTW_N; ++k) {
        __syncthreads();  // also orders the LDS preload before first use
#pragma unroll
        for (int c = 0; c < 3; ++c) {
            const int i = rows[c];
            if (i > DTW_M) continue;                 // row doesn't exist
            if (i > k || (k - i) > DTW_N) continue;  // cell not on this diagonal
            float v;
            if (i == 0) {
                v = (k == 0) ? 0.0f : INFV;          // D[0,0]=0, D[0,j>0]=INF
            } else {
                // neighbor's value on diagonal k-1 (row i-1); always valid here
                const float nbr_new = dbuf[(k + 1) & 1][i - 1];
                if (k == i) {
                    v = INFV;                        // j == 0 border: D[i,0]=INF
                } else {
                    const float d = ss[c] - lt[k - i - 1];
                    const float m = fminf(fminf(my_prev[c], nbr_new), nbr_prev[c]);
                    v = d * d + m;
                }
                nbr_prev[c] = nbr_new;               // becomes upleft next diagonal
            }
            dbuf[k & 1][i] = v;
            my_prev[c]     = v;
            if (i == DTW_M) result = v;              // only tid 0 (c==2) hits this
        }
    }

    if (tid == 0) out[0] = sqrtf(result);
}

extern "C" void kernel_launch(void* const* d_in, const int* in_sizes, int n_in,
                              void* d_out, int out_size, void* d_ws, size_t ws_size,
                              hipStream_t stream) {
    (void)in_sizes; (void)n_in; (void)out_size; (void)d_ws; (void)ws_size;
    const float* src = (const float*)d_in[0];  // source, 2048 f32
    const float* tgt = (const float*)d_in[1];  // target, 2048 f32
    float* out = (float*)d_out;                // 1 f32
    dtw_wavefront_kernel<<<1, NTHREADS, 0, stream>>>(src, tgt, out);
}